// DTW_Loss_40845138985586
// MI455X (gfx1250) — compile-verified
//
#include <hip/hip_runtime.h>
#include <hip/hip_bf16.h>

typedef __attribute__((ext_vector_type(2))) float v2f;
typedef __attribute__((ext_vector_type(8))) float v8f;

#define BLOCK 256
#define GRIDX 8

// Kernel 1: per-batch-row gather + |diff| accumulation, WMMA wave reduction,
// one fp32 partial per block written to workspace (no atomics).
__global__ __launch_bounds__(BLOCK) void dtw_partial_kernel(
    const float* __restrict__ preds, const float* __restrict__ targets,
    const long long* __restrict__ path_i, const long long* __restrict__ path_j,
    float* __restrict__ partials, int S, int P)
{
    const int b = blockIdx.y;
    const float2* __restrict__ pr = reinterpret_cast<const float2*>(preds)   + (size_t)b * S;
    const float2* __restrict__ tg = reinterpret_cast<const float2*>(targets) + (size_t)b * S;
    const long long* __restrict__ pi = path_i + (size_t)b * P;
    const long long* __restrict__ pj = path_j + (size_t)b * P;

    const int stride = gridDim.x * BLOCK;
    float acc0 = 0.0f, acc1 = 0.0f;

    for (int p = blockIdx.x * BLOCK + threadIdx.x; p < P; p += stride) {
        // Prefetch the next chunk of the index streams (speculative; safe OOB).
        __builtin_prefetch(pi + p + stride, 0, 0);
        __builtin_prefetch(pj + p + stride, 0, 0);

        const int i = (int)pi[p];
        const int j = (int)pj[p];
        const float2 pv = pr[i];
        const float2 tv = tg[j];
        acc0 += fabsf(pv.x - tv.x);
        acc1 += fabsf(pv.y - tv.y);
    }

    // --- Wave-level reduction of all 64 accumulators (2 per lane x 32 lanes)
    // with a single V_WMMA_F32_16X16X4_F32 against an all-ones B matrix.
    // A(16x4) layout: lane m (0..15) holds A[m,0]=acc0, A[m,1]=acc1;
    // lane m+16 holds A[m,2]=acc0, A[m,3]=acc1. With B==1:
    //   D[m,n] = acc0[m]+acc1[m]+acc0[m+16]+acc1[m+16]  (same for every n)
    v2f a;  a[0] = acc0; a[1] = acc1;
    v2f bo; bo[0] = 1.0f; bo[1] = 1.0f;
    v8f c = {};
    v8f d = __builtin_amdgcn_wmma_f32_16x16x4_f32(
        /*neg_a=*/false, a, /*neg_b=*/false, bo,
        /*c_mod=*/(short)0, c, /*reuse_a=*/false, /*reuse_b=*/false);

    // Lanes 0..15 hold D rows M=0..7 in d[0..7]; lanes 16..31 hold M=8..15.
    float s = d[0] + d[1] + d[2] + d[3] + d[4] + d[5] + d[6] + d[7];
    // Combine the two half-wave row groups -> every lane has the full wave sum.
    s += __shfl_xor(s, 16, 32);

    __shared__ float wsum[BLOCK / 32];
    const int lane = threadIdx.x & 31;
    const int wid  = threadIdx.x >> 5;
    if (lane == 0) wsum[wid] = s;
    __syncthreads();
    if (threadIdx.x == 0) {
        float t = 0.0f;
#pragma unroll
        for (int w = 0; w < BLOCK / 32; ++w) t += wsum[w];
        partials[(size_t)blockIdx.y * gridDim.x + blockIdx.x] = t;
    }
}

// Kernel 2: reduce the 2048 block partials, scale by 1/(B*S), write the scalar.
__global__ __launch_bounds__(BLOCK) void dtw_final_kernel(
    const float* __restrict__ partials, float* __restrict__ out,
    int n, float scale)
{
    float s = 0.0f;
    for (int i = threadIdx.x; i < n; i += BLOCK) s += partials[i];

#pragma unroll
    for (int m = 16; m >= 1; m >>= 1) s += __shfl_xor(s, m, 32);

    __shared__ float wsum[BLOCK / 32];
    const int lane = threadIdx.x & 31;
    const int wid  = threadIdx.x >> 5;
    if (lane == 0) wsum[wid] = s;
    __syncthreads();
    if (threadIdx.x == 0) {
        float t = 0.0f;
#pragma unroll
        for (int w = 0; w < BLOCK / 32; ++w) t += wsum[w];
        out[0] = t * scale;   // unconditional overwrite: no zero-init needed
    }
}

extern "C" void kernel_launch(void* const* d_in, const int* in_sizes, int n_in,
                              void* d_out, int out_size, void* d_ws, size_t ws_size,
                              hipStream_t stream) {
    const float*     preds   = (const float*)d_in[0];       // (B, S, 2) f32
    const float*     targets = (const float*)d_in[1];       // (B, S, 2) f32
    const long long* path_i  = (const long long*)d_in[2];   // (B, P) i64
    const long long* path_j  = (const long long*)d_in[3];   // (B, P) i64
    float* out = (float*)d_out;

    const int B = 256;                      // per reference setup
    const int S = in_sizes[0] / (B * 2);    // 8192
    const int P = in_sizes[2] / B;          // 16383

    float* partials = (float*)d_ws;         // GRIDX*B floats = 8 KB scratch

    dim3 grid(GRIDX, B);
    dtw_partial_kernel<<<grid, BLOCK, 0, stream>>>(preds, targets, path_i, path_j,
                                                   partials, S, P);

    const int   n     = GRIDX * B;
    const float scale = 1.0f / ((float)B * (float)S);  // 2^-21, exact
    dtw_final_kernel<<<1, BLOCK, 0, stream>>>(partials, out, n, scale);
}